// SSIMLoss_77567109365946
// MI455X (gfx1250) — compile-verified
//
#include <hip/hip_runtime.h>

typedef float v2f __attribute__((ext_vector_type(2)));
typedef float v8f __attribute__((ext_vector_type(8)));

#define OUTW 492          // 512 - 2*shave, shave = scale(4)+6 = 10
#define TILE 32
#define HALO 42           // TILE + 10 (box radius 5 each side)
#define LDW  44           // padded LDS row stride
#define KCH  7            // ceil(26/4) K-chunks of the 16x16x4 WMMA
#define NB   16           // ceil(492/32) tiles per dimension

__global__ __launch_bounds__(128)
void ssim_tile_kernel(const float* __restrict__ srp, const float* __restrict__ hrp,
                      float* __restrict__ partial)
{
    __shared__ float ySr[HALO * LDW];
    __shared__ float yHr[HALO * LDW];
    __shared__ float Tq[4][28 * 16];   // per-wave stage-1 result (K=28 rows x N=16)
    __shared__ float red[128];

    const int tid = threadIdx.x;
    const int img = blockIdx.z;
    const int R0  = blockIdx.y * TILE;
    const int C0  = blockIdx.x * TILE;

    const float w0 = 65.738f  / 256.0f;
    const float w1 = 129.057f / 256.0f;
    const float w2 = 25.064f  / 256.0f;
    const float inv255 = 1.0f / 255.0f;

    // ---- cooperative halo fill: luma of clipped+shaved inputs, zero pad at borders ----
    for (int p = tid; p < HALO * HALO; p += 128) {
        const int hrow = p / HALO;
        const int hcol = p - hrow * HALO;
        const int orow = R0 + hrow - 5;
        const int ocol = C0 + hcol - 5;
        float ys = 0.0f, yh = 0.0f;
        if (orow >= 0 && orow < OUTW && ocol >= 0 && ocol < OUTW) {
            const int base = (img * 3 * 512 + (orow + 10)) * 512 + (ocol + 10);
            float a0 = fminf(fmaxf(srp[base         ] * inv255, 0.f), 1.f);
            float a1 = fminf(fmaxf(srp[base + 262144] * inv255, 0.f), 1.f);
            float a2 = fminf(fmaxf(srp[base + 524288] * inv255, 0.f), 1.f);
            float b0 = fminf(fmaxf(hrp[base         ] * inv255, 0.f), 1.f);
            float b1 = fminf(fmaxf(hrp[base + 262144] * inv255, 0.f), 1.f);
            float b2 = fminf(fmaxf(hrp[base + 524288] * inv255, 0.f), 1.f);
            ys = w0 * a0 + w1 * a1 + w2 * a2;
            yh = w0 * b0 + w1 * b1 + w2 * b2;
        }
        ySr[hrow * LDW + hcol] = ys;
        yHr[hrow * LDW + hcol] = yh;
    }
    __syncthreads();

    const int wave = tid >> 5;
    const int lane = tid & 31;
    const int grp  = lane >> 4;        // half-wave group (K/M-high select)
    const int ln   = lane & 15;        // N index (and M-local for A rows)
    const int subR = (wave >> 1) * 16; // subtile origin inside tile
    const int subC = (wave & 1) * 16;
    float* T = Tq[wave];

    v8f box[5];

#pragma unroll
    for (int q = 0; q < 5; ++q) {
        // ---- Stage 1: T(26x16) = Yh(26x26) * H(26x16), H banded 0/1 ----
#pragma unroll
        for (int mt = 0; mt < 2; ++mt) {
            v8f acc = {};
#pragma unroll
            for (int k = 0; k < KCH; ++k) {
                const int i  = mt * 16 + ln;       // A-matrix row (halo-local)
                const int jx = 4 * k + 2 * grp;    // A-matrix cols for this lane
                const int jy = jx + 1;
                const bool vi = (i < 26);
                const int rowb = (subR + i) * LDW + subC;
                float sx = (vi && jx < 26) ? ySr[rowb + jx] : 0.0f;
                float sy = (vi && jy < 26) ? ySr[rowb + jy] : 0.0f;
                float tx = (vi && jx < 26) ? yHr[rowb + jx] : 0.0f;
                float ty = (vi && jy < 26) ? yHr[rowb + jy] : 0.0f;
                float ax, ay;
                if      (q == 0) { ax = sx;      ay = sy;      }
                else if (q == 1) { ax = tx;      ay = ty;      }
                else if (q == 2) { ax = sx * sx; ay = sy * sy; }
                else if (q == 3) { ax = tx * tx; ay = ty * ty; }
                else             { ax = sx * tx; ay = sy * ty; }
                v2f a; a.x = ax; a.y = ay;
                v2f b;   // H[j][n] = 1 iff n <= j <= n+10 (band of width 11)
                b.x = (ln <= jx && jx <= ln + 10) ? 1.0f : 0.0f;
                b.y = (ln <= jy && jy <= ln + 10) ? 1.0f : 0.0f;
                acc = __builtin_amdgcn_wmma_f32_16x16x4_f32(
                          false, a, false, b, (short)0, acc, false, false);
            }
#pragma unroll
            for (int v = 0; v < 8; ++v) {          // D-frag row = mt*16 + v + 8*grp
                const int row = mt * 16 + v + 8 * grp;
                if (row < 28) T[row * 16 + ln] = acc[v];
            }
        }
        __syncthreads();

        // ---- Stage 2: Out(16x16) = V(16x28) * T(28x16), V banded 0/1 ----
        v8f acc2 = {};
#pragma unroll
        for (int k = 0; k < KCH; ++k) {
            const int ix = 4 * k + 2 * grp;
            const int iy = ix + 1;
            v2f a;   // V[m][i] = 1 iff m <= i <= m+10
            a.x = (ln <= ix && ix <= ln + 10) ? 1.0f : 0.0f;
            a.y = (ln <= iy && iy <= ln + 10) ? 1.0f : 0.0f;
            v2f b; b.x = T[ix * 16 + ln]; b.y = T[iy * 16 + ln];
            acc2 = __builtin_amdgcn_wmma_f32_16x16x4_f32(
                       false, a, false, b, (short)0, acc2, false, false);
        }
        box[q] = acc2;
        __syncthreads();
    }

    // ---- SSIM map + per-block reduction ----
    const float C1c = 6.5025f;     // (0.01*255)^2
    const float C2c = 58.5225f;    // (0.03*255)^2
    const float inv121 = 1.0f / 121.0f;

    float lsum = 0.0f;
#pragma unroll
    for (int v = 0; v < 8; ++v) {
        const int orow = R0 + subR + v + 8 * grp;
        const int ocol = C0 + subC + ln;
        const float mu1  = box[0][v] * inv121;
        const float mu2  = box[1][v] * inv121;
        const float mu1s = mu1 * mu1, mu2s = mu2 * mu2, mu12 = mu1 * mu2;
        const float s1   = box[2][v] * inv121 - mu1s;
        const float s2   = box[3][v] * inv121 - mu2s;
        const float s12  = box[4][v] * inv121 - mu12;
        const float num  = (2.0f * mu12 + C1c) * (2.0f * s12 + C2c);
        const float den  = (mu1s + mu2s + C1c) * (s1 + s2 + C2c);
        const float term = 1.0f - num / den;
        lsum += (orow < OUTW && ocol < OUTW) ? term : 0.0f;
    }
    red[tid] = lsum;
    __syncthreads();
#pragma unroll
    for (int s = 64; s > 0; s >>= 1) {
        if (tid < s) red[tid] += red[tid + s];
        __syncthreads();
    }
    if (tid == 0)
        partial[(blockIdx.z * gridDim.y + blockIdx.y) * gridDim.x + blockIdx.x] = red[0];
}

// Deterministic final reduction (fixed partition + tree; no float atomics)
__global__ __launch_bounds__(256)
void ssim_reduce_kernel(const float* __restrict__ partial, int n, float* __restrict__ out)
{
    __shared__ float red[256];
    float s = 0.0f;
    for (int i = threadIdx.x; i < n; i += 256) s += partial[i];
    red[threadIdx.x] = s;
    __syncthreads();
    for (int k = 128; k > 0; k >>= 1) {
        if (threadIdx.x < k) red[threadIdx.x] += red[threadIdx.x + k];
        __syncthreads();
    }
    if (threadIdx.x == 0) out[0] = red[0] * (1.0f / 16.0f);   // / batch_size
}

extern "C" void kernel_launch(void* const* d_in, const int* in_sizes, int n_in,
                              void* d_out, int out_size, void* d_ws, size_t ws_size,
                              hipStream_t stream)
{
    const float* sr = (const float*)d_in[0];
    const float* hr = (const float*)d_in[1];
    float* partial  = (float*)d_ws;            // NB*NB*16 = 4096 floats (16 KB)

    dim3 grid(NB, NB, 16);
    ssim_tile_kernel<<<grid, 128, 0, stream>>>(sr, hr, partial);
    ssim_reduce_kernel<<<1, 256, 0, stream>>>(partial, NB * NB * 16, (float*)d_out);
}